// ECELoss_49813030699083
// MI455X (gfx1250) — compile-verified
//
#include <hip/hip_runtime.h>
#include <stdint.h>

// ECE/MCE over [N,64] logits. Memory-bound: 1.024 GB logits @ 23.3 TB/s ~ 44us floor.
// gfx1250 path: async global->LDS streaming (ASYNCcnt) + wave32 shfl reductions.

#define NBINS            15
#define ROWS_PER_CHUNK   16                       // 16 rows * 256B = 4KB per wave-buffer
#define CHUNK_FLOATS     (ROWS_PER_CHUNK * 64)    // 1024 floats
#define CHUNK_BYTES      (CHUNK_FLOATS * 4)       // 4096 bytes
#define WAVES_PER_BLOCK  8
#define TPB              (WAVES_PER_BLOCK * 32)
#define GRID_BLOCKS      2048

__global__ void ece_init_kernel(unsigned long long* gConf, unsigned* gCnt, unsigned* gAcc) {
    int t = threadIdx.x;
    if (t < NBINS) { gConf[t] = 0ull; gCnt[t] = 0u; gAcc[t] = 0u; }
}

__global__ __launch_bounds__(TPB)
void ece_main_kernel(const float* __restrict__ logits,
                     const int*   __restrict__ labels,
                     const float* __restrict__ temperature,
                     int nRows,
                     unsigned long long* __restrict__ gConf,
                     unsigned* __restrict__ gCnt,
                     unsigned* __restrict__ gAcc) {
    // Double-buffered staging tiles: 2 * 8 waves * 4KB = 64KB LDS (WGP has 320KB).
    __shared__ float tile[2][WAVES_PER_BLOCK][CHUNK_FLOATS];
    __shared__ unsigned long long sConf[NBINS];
    __shared__ unsigned sCnt[NBINS];
    __shared__ unsigned sAcc[NBINS];

    const int tid  = threadIdx.x;
    const int lane = tid & 31;
    const int wib  = tid >> 5;  // wave in block

    if (tid < NBINS) { sConf[tid] = 0ull; sCnt[tid] = 0u; sAcc[tid] = 0u; }
    __syncthreads();

    const float invT = 1.0f / temperature[0];
    const float kexp = invT * 1.44269504088896340736f;   // log2(e)/T -> exp2f == v_exp_f32

    const int totalChunks = (nRows + ROWS_PER_CHUNK - 1) / ROWS_PER_CHUNK;
    const int waveGlobal  = (int)blockIdx.x * WAVES_PER_BLOCK + wib;
    const int waveStride  = (int)gridDim.x * WAVES_PER_BLOCK;

    // Flat shared address low 32 bits == workgroup-relative LDS byte offset (aperture rule).
    const unsigned ldsBase[2] = {
        (unsigned)(uintptr_t)(&tile[0][wib][0]),
        (unsigned)(uintptr_t)(&tile[1][wib][0])
    };

    // Issue one 4KB chunk as 8 async B128 transfers (32 lanes * 16B = 512B each).
    auto issue_chunk = [&](int chunk, unsigned lbase) {
        long long rowStart = (long long)chunk * ROWS_PER_CHUNK;
        long long remRows  = (long long)nRows - rowStart;
        if (remRows > ROWS_PER_CHUNK) remRows = ROWS_PER_CHUNK;
        const unsigned remBytes = (unsigned)(remRows * 256ll);
        const float* src = logits + rowStart * 64;   // uniform -> SGPR pair (GVS saddr)
        const unsigned laneOff = (unsigned)lane * 16u;
#pragma unroll
        for (int i = 0; i < 8; ++i) {
            unsigned voff  = laneOff + (unsigned)i * 512u;
            unsigned gvoff = (voff < remBytes) ? voff : (remBytes - 16u); // clamp tail in-bounds
            unsigned laddr = lbase + voff;
            asm volatile("global_load_async_to_lds_b128 %0, %1, %2"
                         :
                         : "v"(laddr), "v"(gvoff), "s"(src)
                         : "memory");
        }
    };

    int chunk = waveGlobal;
    int buf = 0;
    if (chunk < totalChunks) issue_chunk(chunk, ldsBase[0]);

    for (; chunk < totalChunks; chunk += waveStride) {
        const int next = chunk + waveStride;
        if (next < totalChunks) {
            issue_chunk(next, ldsBase[buf ^ 1]);
            // 8 in-flight belong to 'next'; loads complete in order -> current chunk done.
            asm volatile("s_wait_asynccnt 0x8" ::: "memory");
        } else {
            asm volatile("s_wait_asynccnt 0x0" ::: "memory");
        }

        const float* tp = &tile[buf][wib][0];
#pragma unroll 4
        for (int r = 0; r < ROWS_PER_CHUNK; ++r) {
            long long row = (long long)chunk * ROWS_PER_CHUNK + r;
            if (row >= nRows) break;

            // lane L covers classes 2L, 2L+1 (ds_load_b64, conflict-free).
            float2 v = *(const float2*)(tp + r * 64 + lane * 2);

            // max + argmax (first occurrence on tie)
            float m = v.x; int mi = lane * 2;
            if (v.y > m) { m = v.y; mi = lane * 2 + 1; }
#pragma unroll
            for (int off = 16; off >= 1; off >>= 1) {
                float om = __shfl_xor(m, off, 32);
                int   oi = __shfl_xor(mi, off, 32);
                if (om > m || (om == m && oi < mi)) { m = om; mi = oi; }
            }

            // sum exp((l-max)/T); confidence = 1/sum
            float e = exp2f((v.x - m) * kexp) + exp2f((v.y - m) * kexp);
#pragma unroll
            for (int off = 16; off >= 1; off >>= 1) e += __shfl_xor(e, off, 32);

            if (lane == 0) {
                float conf = 1.0f / e;
                int b = (int)ceilf(conf * 15.0f) - 1;
                b = b < 0 ? 0 : (b > NBINS - 1 ? NBINS - 1 : b);
                atomicAdd(&sCnt[b], 1u);
                if (mi == labels[row]) atomicAdd(&sAcc[b], 1u);
                // Q2.30 fixed point: deterministic, order-independent accumulation
                atomicAdd(&sConf[b], (unsigned long long)(conf * 1073741824.0f));
            }
        }
        buf ^= 1;
    }

    __syncthreads();
    if (tid < NBINS) {
        if (sCnt[tid])  atomicAdd(&gCnt[tid],  sCnt[tid]);
        if (sAcc[tid])  atomicAdd(&gAcc[tid],  sAcc[tid]);
        if (sConf[tid]) atomicAdd(&gConf[tid], sConf[tid]);
    }
}

__global__ void ece_final_kernel(const unsigned long long* __restrict__ gConf,
                                 const unsigned* __restrict__ gCnt,
                                 const unsigned* __restrict__ gAcc,
                                 int nRows, float* __restrict__ out) {
    int lane = threadIdx.x & 31;
    float gap = 0.0f, contrib = 0.0f;
    if (lane < NBINS) {
        unsigned c = gCnt[lane];
        if (c > 0u) {
            float cf      = (float)c;
            float confSum = (float)((double)gConf[lane] * (1.0 / 1073741824.0));
            float accSum  = (float)gAcc[lane];
            gap     = fabsf(confSum / cf - accSum / cf);
            contrib = gap * (cf / (float)nRows);
        }
    }
#pragma unroll
    for (int off = 16; off >= 1; off >>= 1) {
        contrib += __shfl_xor(contrib, off, 32);
        float og = __shfl_xor(gap, off, 32);
        gap = og > gap ? og : gap;
    }
    if (lane == 0) { out[0] = contrib; out[1] = gap; }
}

extern "C" void kernel_launch(void* const* d_in, const int* in_sizes, int n_in,
                              void* d_out, int out_size, void* d_ws, size_t ws_size,
                              hipStream_t stream) {
    const float* logits      = (const float*)d_in[0];
    const int*   labels      = (const int*)d_in[1];
    const float* temperature = (const float*)d_in[2];
    const int    nRows       = in_sizes[1];       // labels count == N

    unsigned long long* gConf = (unsigned long long*)d_ws;
    unsigned* gCnt = (unsigned*)(gConf + NBINS);
    unsigned* gAcc = gCnt + NBINS;
    float* out = (float*)d_out;

    hipLaunchKernelGGL(ece_init_kernel, dim3(1), dim3(64), 0, stream, gConf, gCnt, gAcc);
    hipLaunchKernelGGL(ece_main_kernel, dim3(GRID_BLOCKS), dim3(TPB), 0, stream,
                       logits, labels, temperature, nRows, gConf, gCnt, gAcc);
    hipLaunchKernelGGL(ece_final_kernel, dim3(1), dim3(32), 0, stream,
                       gConf, gCnt, gAcc, nRows, out);
    (void)n_in; (void)out_size; (void)ws_size;
}